// ConnectivityLoss3D_62577673503373
// MI455X (gfx1250) — compile-verified
//
#include <hip/hip_runtime.h>
#include <hip/hip_bf16.h>
#include <stdint.h>

// ---------------------------------------------------------------------------
// ConnectivityLoss3D for MI455X (gfx1250, wave32).
//
// Shapes: x,y = [4,1,192,192,192] f32; is_gt = [4] f32; out = [1] f32.
//
// Roofline: 226 MB of x/y reads -> ~9.7us floor @ 23.3 TB/s; everything else
// (3.5 MB bit-plane, partials) is L2-resident. conn term is reduced to a
// 26-neighbor periodic NOR over packed pred bits (acc==pred  <=>  all 26
// neighbors are 0). CDNA5 paths: wave32 ballot bit-packing, Tensor Data Mover
// staging of bit-slices into LDS (TENSORcnt), global_prefetch_b8 on streams.
// ---------------------------------------------------------------------------

typedef unsigned int u32;
typedef unsigned long long u64;

#define DIMB 4
#define DIMD 192
#define DIMH 192
#define DIMW 192
#define W32 6                       // 192 bits per row = 6 x u32 (exact)
#define SLICE_WORDS (DIMH * W32)    // 1152 u32 per (b,d) bit-slice
#define SLICE_VOX (DIMH * DIMW)     // 36864 voxels per slice
#define NSLICE (DIMB * DIMD)        // 768 blocks
#define BITS_BYTES ((size_t)NSLICE * SLICE_WORDS * sizeof(u32))  // 3538944 B

#ifndef __has_builtin
#define __has_builtin(x) 0
#endif

#if defined(__AMDGCN__) && __has_builtin(__builtin_amdgcn_tensor_load_to_lds) && \
    __has_builtin(__builtin_amdgcn_s_wait_tensorcnt)
#define USE_TDM 1
#else
#define USE_TDM 0
#endif

typedef unsigned int v4u __attribute__((ext_vector_type(4)));
typedef int          v4i __attribute__((ext_vector_type(4)));
typedef int          v8i __attribute__((ext_vector_type(8)));

// ---------------------------------------------------------------------------
// Kernel: connectivity = popcount of voxels whose 26 periodic neighbors are
// all zero. Rows are exactly 192 bits = 6 u32, so word-index wrap (mod 6)
// implements the periodic wrap in W for free. Placed FIRST in the file so the
// disasm snippet shows the tensor_load_to_lds / s_wait_tensorcnt lowering.
// ---------------------------------------------------------------------------
__device__ __forceinline__ u32 row_or3(const u32* s, int h, int i, int im1, int ip1) {
    const u32* r = s + h * W32;
    u32 c = r[i], lw = r[im1], rw = r[ip1];
    return c | ((c << 1) | (lw >> 31)) | ((c >> 1) | (rw << 31));
}
__device__ __forceinline__ u32 row_shlr(const u32* s, int h, int i, int im1, int ip1) {
    const u32* r = s + h * W32;
    u32 c = r[i], lw = r[im1], rw = r[ip1];
    return ((c << 1) | (lw >> 31)) | ((c >> 1) | (rw << 31));   // exclude self
}

__global__ void __launch_bounds__(256)
k_conn(const u32* __restrict__ bits, u32* __restrict__ connPartial) {
    const int bd = blockIdx.x;
    const int b = bd / DIMD;
    const int d = bd % DIMD;
    const int dzs[3] = { (d + DIMD - 1) % DIMD, d, (d + 1) % DIMD };

    __shared__ u32 sb[3][SLICE_WORDS];                 // 13.8 KB of 320 KB LDS
    const int tid = threadIdx.x;

#if USE_TDM
    // Scalar (SGPR) branch: readfirstlane makes the condition provably
    // wave-uniform -> s_cmp + s_cbranch, so waves 1..7 cannot reach the TDM
    // issue (TDM ignores EXEC, so an exec-mask-only guard is not safe).
    if (__builtin_amdgcn_readfirstlane((int)(threadIdx.x >> 5)) == 0) {
        #pragma unroll
        for (int z = 0; z < 3; ++z) {
            const u32* src = bits + ((size_t)b * DIMD + dzs[z]) * SLICE_WORDS;
            const u64 gaddr  = (u64)(uintptr_t)src;
            const u32 ldsAddr = (u32)(uintptr_t)&sb[z][0];   // low 32 = LDS offset

            v4u g0;
            g0[0] = 1u;                                  // count=1, user mode
            g0[1] = ldsAddr;                             // lds_addr (bytes)
            g0[2] = (u32)gaddr;                          // global_addr[31:0]
            g0[3] = ((u32)(gaddr >> 32) & 0x01FFFFFFu) | (2u << 30); // addr[56:32]|type=2

            v8i g1;
            g1[0] = 0x00020000;                          // data_size = 4B
            g1[1] = (int)((SLICE_WORDS & 0xFFFFu) << 16);   // tensor_dim0 lo
            g1[2] = (int)((SLICE_WORDS >> 16) | (1u << 16)); // dim0 hi | tensor_dim1=1
            g1[3] = (int)(SLICE_WORDS << 16);            // tile_dim0 = 1152
            g1[4] = 1;                                   // tile_dim1=1, tile_dim2=0
            g1[5] = (int)SLICE_WORDS;                    // tensor_dim0_stride
            g1[6] = (int)((u32)SLICE_WORDS << 16);       // tensor_dim1_stride lo
            g1[7] = 0;

            v4i gz = {0, 0, 0, 0};
#if __clang_major__ >= 23
            v8i gz8 = {0, 0, 0, 0, 0, 0, 0, 0};
            __builtin_amdgcn_tensor_load_to_lds(g0, g1, gz, gz, gz8, 0);
#else
            __builtin_amdgcn_tensor_load_to_lds(g0, g1, gz, gz, 0);
#endif
        }
        __builtin_amdgcn_s_wait_tensorcnt(0);            // all 3 DMAs complete
    }
#else
    {
        u32* flat = &sb[0][0];
        for (int t = tid; t < 3 * SLICE_WORDS; t += 256) {
            const int z = t / SLICE_WORDS, r = t % SLICE_WORDS;
            flat[t] = bits[((size_t)b * DIMD + dzs[z]) * SLICE_WORDS + r];
        }
    }
#endif
    __syncthreads();

    int cnt = 0;
    for (int t = tid; t < SLICE_WORDS; t += 256) {
        const int h = t / W32, i = t % W32;
        const int im1 = (i + W32 - 1) % W32, ip1 = (i + 1) % W32;
        const int hm = (h + DIMH - 1) % DIMH, hp = (h + 1) % DIMH;
        u32 o = 0;
        #pragma unroll
        for (int z = 0; z < 3; ++z) {
            const u32* s = sb[z];
            o |= row_or3(s, hm, i, im1, ip1);
            o |= (z == 1) ? row_shlr(s, h, i, im1, ip1)   // center: exclude self
                          : row_or3 (s, h, i, im1, ip1);
            o |= row_or3(s, hp, i, im1, ip1);
        }
        cnt += __popc(~o);                              // all 32 bits valid (192=6*32)
    }

    __shared__ int redi[256];
    redi[tid] = cnt;
    __syncthreads();
    for (int s = 128; s > 0; s >>= 1) {
        if (tid < s) redi[tid] += redi[tid + s];
        __syncthreads();
    }
    if (tid == 0) connPartial[bd] = (u32)redi[0];
}

// ---------------------------------------------------------------------------
// Kernel: fused BCE partial-reduction + pred bit-packing (one block per (b,d)
// slice; 256 threads = 8 waves; 144 iterations, unrolled 4x for MLP).
// softplus(-|x|) = log(1+exp(-|x|)) via v_exp_f32/v_log_f32: per-element
// rounding averages out in the 7M-element mean.
// ---------------------------------------------------------------------------
__global__ void __launch_bounds__(256)
k_bce_pack(const float* __restrict__ x, const float* __restrict__ y,
           u32* __restrict__ bits, float* __restrict__ bcePartial) {
    const int bd = blockIdx.x;                         // 0..767 == b*192+d
    const size_t base = (size_t)bd * SLICE_VOX;
    const float* __restrict__ xs = x + base;
    const float* __restrict__ ys = y + base;
    u32* __restrict__ bslice = bits + (size_t)bd * SLICE_WORDS;

    const int tid = threadIdx.x;
    float acc = 0.0f;

    #pragma unroll 4
    for (int l = tid; l < SLICE_VOX; l += 256) {
        __builtin_prefetch(xs + l + 4096, 0, 0);       // global_prefetch_b8
        __builtin_prefetch(ys + l + 4096, 0, 0);
        const float xv = xs[l];
        const float yv = ys[l];
        const float e  = __expf(-fabsf(xv));           // v_exp_f32
        acc += fmaxf(xv, 0.0f) - xv * yv + __logf(1.0f + e);  // v_log_f32
        // pack pred bits: lanes of a wave cover 32 consecutive w positions
#if __has_builtin(__builtin_amdgcn_ballot_w32)
        const u32 m = __builtin_amdgcn_ballot_w32(xv > 0.0f);
#else
        const u32 m = (u32)__ballot(xv > 0.0f);
#endif
        if ((tid & 31) == 0) bslice[l >> 5] = m;
    }

    __shared__ float red[256];
    red[tid] = acc;
    __syncthreads();
    for (int s = 128; s > 0; s >>= 1) {
        if (tid < s) red[tid] += red[tid + s];
        __syncthreads();
    }
    if (tid == 0) bcePartial[bd] = red[0];
}

// ---------------------------------------------------------------------------
// Kernel: fold 768 per-slice partials into the scalar loss (deterministic,
// fixed-order tree; no atomics anywhere).
// ---------------------------------------------------------------------------
__global__ void __launch_bounds__(256)
k_finalize(const float* __restrict__ bcePartial, const u32* __restrict__ connPartial,
           const float* __restrict__ is_gt, float* __restrict__ out) {
    const float invN = 1.0f / (192.0f * 192.0f * 192.0f);
    const int tid = threadIdx.x;
    float acc = 0.0f;
    for (int t = tid; t < NSLICE; t += 256) {
        const int b = t / DIMD;
        acc += is_gt[b] * bcePartial[t] * invN
             + 1e-4f * (float)connPartial[t] * invN;
    }
    __shared__ float red[256];
    red[tid] = acc;
    __syncthreads();
    for (int s = 128; s > 0; s >>= 1) {
        if (tid < s) red[tid] += red[tid + s];
        __syncthreads();
    }
    if (tid == 0) out[0] = red[0];
}

// ---------------------------------------------------------------------------
extern "C" void kernel_launch(void* const* d_in, const int* in_sizes, int n_in,
                              void* d_out, int out_size, void* d_ws, size_t ws_size,
                              hipStream_t stream) {
    const float* x     = (const float*)d_in[0];
    const float* y     = (const float*)d_in[1];
    const float* is_gt = (const float*)d_in[2];
    float* out = (float*)d_out;

    u32*   bits        = (u32*)d_ws;
    float* bcePartial  = (float*)((char*)d_ws + BITS_BYTES);
    u32*   connPartial = (u32*)((char*)d_ws + BITS_BYTES + NSLICE * sizeof(float));

    k_bce_pack<<<NSLICE, 256, 0, stream>>>(x, y, bits, bcePartial);
    k_conn    <<<NSLICE, 256, 0, stream>>>(bits, connPartial);
    k_finalize<<<1, 256, 0, stream>>>(bcePartial, connPartial, is_gt, out);
}